// WindowAttention3D_54339926229586
// MI455X (gfx1250) — compile-verified
//
#include <hip/hip_runtime.h>
#include <hip/hip_bf16.h>
#include <math.h>

typedef float v2f __attribute__((ext_vector_type(2)));
typedef float v8f __attribute__((ext_vector_type(8)));

#define NTOK 392
#define CDIM 128
#define DD 8
#define HH 7
#define WW 7
#define NHEAD 8
#define NPNT 4
#define HDIM 16
#define PI_F 3.14159265358979323846f
#define ROWT 25   // ceil(392/16)

// ---------------------------------------------------------------------------
// K1: diagonal of the position-embed matrix. diag[b][n] with n -> (d,h,w).
// z_e = cumsum_D * m / PI ; y_e = cumsum_H * m / PI ; x_e = cumsum_W * m / PI
// embed_diag = m * (sin(x_e) + sin(PI*y_e) + z_e)
// ---------------------------------------------------------------------------
__global__ __launch_bounds__(128) void k_diag(const float* __restrict__ mask,
                                              float* __restrict__ diag) {
  int b = blockIdx.x;
  for (int n = threadIdx.x; n < NTOK; n += blockDim.x) {
    const float* mb = mask + ((size_t)b * NTOK + n) * (DD * HH * WW);
    int d = n / (HH * WW);
    int rem = n % (HH * WW);
    int h = rem / WW;
    int w = rem % WW;
    float cz = 0.f, cy = 0.f, cx = 0.f;
    for (int i = 0; i <= d; ++i) cz += mb[(i * HH + h) * WW + w];
    for (int i = 0; i <= h; ++i) cy += mb[(d * HH + i) * WW + w];
    for (int i = 0; i <= w; ++i) cx += mb[(d * HH + h) * WW + i];
    float m = mb[(d * HH + h) * WW + w];
    float ze = cz * m / PI_F;
    float ye = cy * m / PI_F;
    float xe = cx * m / PI_F;
    diag[(size_t)b * NTOK + n] = m * (sinf(xe) + sinf(PI_F * ye) + ze);
  }
}

// ---------------------------------------------------------------------------
// K2: fused value / offset / attn-logit GEMMs via V_WMMA_F32_16X16X4_F32.
// One block = (window b, 16-row tile). 8 waves x 2 tiles = 16 col tiles over
// the concatenated output [Wv(128) | Woff(96) | Wattn(32)].
// A fragment: a[i] = A[lane%16][k0 + i + 2*(lane/16)]
// B fragment: b[i] = W[(k0 + i + 2*(lane/16))*ncols + col0 + lane%16]
// D fragment: d[i] = D[i + 8*(lane/16)][col0 + lane%16]
// ---------------------------------------------------------------------------
__global__ __launch_bounds__(256) void k_qkv(
    const float* __restrict__ x, const float* __restrict__ diag,
    const float* __restrict__ wv, const float* __restrict__ bv,
    const float* __restrict__ wo, const float* __restrict__ bo,
    const float* __restrict__ wa, const float* __restrict__ ba,
    float* __restrict__ val, float* __restrict__ off, float* __restrict__ aw) {
  __shared__ float xs[16][CDIM];
  __shared__ float qs[16][CDIM];
  __shared__ float lg[16][32];

  int b = blockIdx.x / ROWT;
  int rt = blockIdx.x % ROWT;
  int row0 = rt * 16;
  int tid = threadIdx.x;

  for (int i = tid; i < 16 * CDIM; i += 256) {
    int r = i >> 7, c = i & 127;
    int gr = row0 + r;
    float v = (gr < NTOK) ? x[((size_t)b * NTOK + gr) * CDIM + c] : 0.f;
    float dg = (gr < NTOK) ? diag[(size_t)b * NTOK + gr] : 0.f;
    xs[r][c] = v;
    qs[r][c] = v + dg;
  }
  __syncthreads();

  int wave = tid >> 5;
  int lane = tid & 31;
  int m = lane & 15;   // M row for A, N col for B/D
  int kh = lane >> 4;  // lane half

  for (int ti = 0; ti < 2; ++ti) {
    int t = wave * 2 + ti;
    const float(*A)[CDIM] = (t < 8) ? xs : qs;
    const float* W;
    const float* Bias;
    int ncols, col0;
    if (t < 8) {
      W = wv; Bias = bv; ncols = 128; col0 = t * 16;
    } else if (t < 14) {
      W = wo; Bias = bo; ncols = 96; col0 = (t - 8) * 16;
    } else {
      W = wa; Bias = ba; ncols = 32; col0 = (t - 14) * 16;
    }
    v8f acc = {};
    for (int k0 = 0; k0 < CDIM; k0 += 4) {
      int ka = k0 + 2 * kh;
      v2f av, bvv;
      av.x = A[m][ka];
      av.y = A[m][ka + 1];
      bvv.x = W[(size_t)ka * ncols + col0 + m];
      bvv.y = W[(size_t)(ka + 1) * ncols + col0 + m];
      acc = __builtin_amdgcn_wmma_f32_16x16x4_f32(false, av, false, bvv,
                                                  (short)0, acc, false, false);
    }
    int n = col0 + m;
    float bias = Bias[n];
#pragma unroll
    for (int i = 0; i < 8; ++i) {
      int r = i + 8 * kh;
      int gr = row0 + r;
      if (t >= 14) {
        lg[r][n] = (gr < NTOK) ? (acc[i] + bias) : 0.f;
      } else if (gr < NTOK) {
        float v = acc[i] + bias;
        if (t < 8)
          val[((size_t)b * NTOK + gr) * CDIM + n] = v;
        else
          off[((size_t)b * NTOK + gr) * (NHEAD * NPNT * 3) + n] = v;
      }
    }
  }
  __syncthreads();

  // softmax over P=4 per (row, head)
  if (tid < 16 * NHEAD) {
    int r = tid >> 3;
    int hh = tid & 7;
    int gr = row0 + r;
    if (gr < NTOK) {
      float l0 = lg[r][hh * 4 + 0];
      float l1 = lg[r][hh * 4 + 1];
      float l2 = lg[r][hh * 4 + 2];
      float l3 = lg[r][hh * 4 + 3];
      float mx = fmaxf(fmaxf(l0, l1), fmaxf(l2, l3));
      float e0 = expf(l0 - mx), e1 = expf(l1 - mx);
      float e2 = expf(l2 - mx), e3 = expf(l3 - mx);
      float inv = 1.0f / (e0 + e1 + e2 + e3);
      float* ab = aw + (((size_t)b * NTOK + gr) * NHEAD + hh) * NPNT;
      ab[0] = e0 * inv; ab[1] = e1 * inv; ab[2] = e2 * inv; ab[3] = e3 * inv;
    }
  }
}

// ---------------------------------------------------------------------------
// K3: trilinear deformable sampling. Block = (window b, head h).
// Per-head value slice (392 x 16 = 25 KB) lives in LDS.
// loc math reduces to px = w + off_x ; py = h + off_y ; pz = d + off_z.
// ---------------------------------------------------------------------------
__global__ __launch_bounds__(256) void k_sample(
    const float* __restrict__ val, const float* __restrict__ off,
    const float* __restrict__ aw, float* __restrict__ sampled) {
  __shared__ float vh[NTOK * HDIM];
  int b = blockIdx.x >> 3;
  int h = blockIdx.x & 7;
  int tid = threadIdx.x;

  for (int i = tid; i < NTOK * HDIM; i += 256)
    vh[i] = val[((size_t)b * NTOK + (i >> 4)) * CDIM + h * HDIM + (i & 15)];
  __syncthreads();

  for (int t = tid; t < NTOK; t += 256) {
    const float* ob = off + (((size_t)b * NTOK + t) * NHEAD + h) * NPNT * 3;
    const float* ab = aw + (((size_t)b * NTOK + t) * NHEAD + h) * NPNT;
    int d = t / (HH * WW);
    int rem = t % (HH * WW);
    int hy = rem / WW;
    int wx = rem % WW;
    float acc[HDIM];
#pragma unroll
    for (int e = 0; e < HDIM; ++e) acc[e] = 0.f;

    for (int p = 0; p < NPNT; ++p) {
      float px = (float)wx + ob[p * 3 + 0];
      float py = (float)hy + ob[p * 3 + 1];
      float pz = (float)d + ob[p * 3 + 2];
      float a = ab[p];
      float x0 = floorf(px), y0 = floorf(py), z0 = floorf(pz);
      float fx = px - x0, fy = py - y0, fz = pz - z0;
      int xi0 = (int)x0, yi0 = (int)y0, zi0 = (int)z0;
#pragma unroll
      for (int dz = 0; dz < 2; ++dz) {
#pragma unroll
        for (int dy = 0; dy < 2; ++dy) {
#pragma unroll
          for (int dx = 0; dx < 2; ++dx) {
            int xi = xi0 + dx, yi = yi0 + dy, zi = zi0 + dz;
            if (xi >= 0 && xi < WW && yi >= 0 && yi < HH && zi >= 0 && zi < DD) {
              float w = (dx ? fx : 1.f - fx) * (dy ? fy : 1.f - fy) *
                        (dz ? fz : 1.f - fz) * a;
              const float* v = &vh[((zi * HH + yi) * WW + xi) * HDIM];
#pragma unroll
              for (int e = 0; e < HDIM; ++e) acc[e] += w * v[e];
            }
          }
        }
      }
    }
    float* o = sampled + ((size_t)b * NTOK + t) * CDIM + h * HDIM;
#pragma unroll
    for (int e = 0; e < HDIM; ++e) o[e] = acc[e];
  }
}

// ---------------------------------------------------------------------------
// K4: sampled @ Wout + bout + x -> LayerNorm -> @ Wproj + bproj -> d_out.
// Two WMMA GEMMs fused around an in-LDS LayerNorm.
// ---------------------------------------------------------------------------
__global__ __launch_bounds__(256) void k_out(
    const float* __restrict__ sampled, const float* __restrict__ x,
    const float* __restrict__ wout, const float* __restrict__ bout,
    const float* __restrict__ wproj, const float* __restrict__ bproj,
    const float* __restrict__ lng, const float* __restrict__ lnb,
    float* __restrict__ out) {
  __shared__ float s_in[16][CDIM];
  __shared__ float s_mid[16][CDIM];
  __shared__ float s_mu[16];
  __shared__ float s_rs[16];

  int b = blockIdx.x / ROWT;
  int rt = blockIdx.x % ROWT;
  int row0 = rt * 16;
  int tid = threadIdx.x;

  for (int i = tid; i < 16 * CDIM; i += 256) {
    int r = i >> 7, c = i & 127;
    int gr = row0 + r;
    s_in[r][c] = (gr < NTOK) ? sampled[((size_t)b * NTOK + gr) * CDIM + c] : 0.f;
  }
  __syncthreads();

  int wave = tid >> 5;
  int lane = tid & 31;
  int m = lane & 15;
  int kh = lane >> 4;
  int col0 = wave * 16;
  int n = col0 + m;

  // GEMM 1: attn = sampled @ wout + bout ; s_mid = x + attn
  {
    v8f acc = {};
    for (int k0 = 0; k0 < CDIM; k0 += 4) {
      int ka = k0 + 2 * kh;
      v2f av, bvv;
      av.x = s_in[m][ka];
      av.y = s_in[m][ka + 1];
      bvv.x = wout[(size_t)ka * CDIM + n];
      bvv.y = wout[(size_t)(ka + 1) * CDIM + n];
      acc = __builtin_amdgcn_wmma_f32_16x16x4_f32(false, av, false, bvv,
                                                  (short)0, acc, false, false);
    }
    float bias = bout[n];
#pragma unroll
    for (int i = 0; i < 8; ++i) {
      int r = i + 8 * kh;
      int gr = row0 + r;
      float resid = (gr < NTOK) ? x[((size_t)b * NTOK + gr) * CDIM + n] : 0.f;
      s_mid[r][n] = acc[i] + bias + resid;
    }
  }
  __syncthreads();

  // LayerNorm stats per row
  if (tid < 16) {
    float mu = 0.f;
    for (int c = 0; c < CDIM; ++c) mu += s_mid[tid][c];
    mu *= (1.0f / CDIM);
    float var = 0.f;
    for (int c = 0; c < CDIM; ++c) {
      float dd = s_mid[tid][c] - mu;
      var += dd * dd;
    }
    var *= (1.0f / CDIM);
    s_mu[tid] = mu;
    s_rs[tid] = rsqrtf(var + 1e-5f);
  }
  __syncthreads();

  for (int i = tid; i < 16 * CDIM; i += 256) {
    int r = i >> 7, c = i & 127;
    s_mid[r][c] = (s_mid[r][c] - s_mu[r]) * s_rs[r] * lng[c] + lnb[c];
  }
  __syncthreads();

  // GEMM 2: out = normed @ wproj + bproj
  {
    v8f acc = {};
    for (int k0 = 0; k0 < CDIM; k0 += 4) {
      int ka = k0 + 2 * kh;
      v2f av, bvv;
      av.x = s_mid[m][ka];
      av.y = s_mid[m][ka + 1];
      bvv.x = wproj[(size_t)ka * CDIM + n];
      bvv.y = wproj[(size_t)(ka + 1) * CDIM + n];
      acc = __builtin_amdgcn_wmma_f32_16x16x4_f32(false, av, false, bvv,
                                                  (short)0, acc, false, false);
    }
    float bias = bproj[n];
#pragma unroll
    for (int i = 0; i < 8; ++i) {
      int r = i + 8 * kh;
      int gr = row0 + r;
      if (gr < NTOK) out[((size_t)b * NTOK + gr) * CDIM + n] = acc[i] + bias;
    }
  }
}

// ---------------------------------------------------------------------------
extern "C" void kernel_launch(void* const* d_in, const int* in_sizes, int n_in,
                              void* d_out, int out_size, void* d_ws,
                              size_t ws_size, hipStream_t stream) {
  const float* x = (const float*)d_in[0];
  const float* mask = (const float*)d_in[1];
  const float* w_value = (const float*)d_in[2];
  const float* b_value = (const float*)d_in[3];
  const float* w_off = (const float*)d_in[4];
  const float* b_off = (const float*)d_in[5];
  const float* w_attn = (const float*)d_in[6];
  const float* b_attn = (const float*)d_in[7];
  const float* w_out = (const float*)d_in[8];
  const float* b_out = (const float*)d_in[9];
  const float* w_proj = (const float*)d_in[10];
  const float* b_proj = (const float*)d_in[11];
  const float* ln_g = (const float*)d_in[12];
  const float* ln_b = (const float*)d_in[13];

  const int Bt = 256;
  float* ws = (float*)d_ws;
  float* diag = ws;                                    // 256*392
  float* val = diag + (size_t)Bt * NTOK;               // 256*392*128
  float* off = val + (size_t)Bt * NTOK * CDIM;         // 256*392*96
  float* aw = off + (size_t)Bt * NTOK * NHEAD * NPNT * 3;  // 256*392*32
  float* sampled = aw + (size_t)Bt * NTOK * NHEAD * NPNT;  // 256*392*128

  k_diag<<<Bt, 128, 0, stream>>>(mask, diag);
  k_qkv<<<Bt * ROWT, 256, 0, stream>>>(x, diag, w_value, b_value, w_off, b_off,
                                       w_attn, b_attn, val, off, aw);
  k_sample<<<Bt * NHEAD, 256, 0, stream>>>(val, off, aw, sampled);
  k_out<<<Bt * ROWT, 256, 0, stream>>>(sampled, x, w_out, b_out, w_proj,
                                       b_proj, ln_g, ln_b, (float*)d_out);
}